// CorrelationLayer1D_13580686590324
// MI455X (gfx1250) — compile-verified
//
#include <hip/hip_runtime.h>

typedef __attribute__((ext_vector_type(2))) float v2f;
typedef __attribute__((ext_vector_type(8))) float v8f;

#define BDIM   4
#define CDIM   128
#define HDIM   160
#define WDIM   320
#define HWSZ   (HDIM * WDIM)      // 51200
#define ND     41                 // displacements 0..40
#define WTILES (WDIM / 16)        // 20

// One wave32 per 16-wide output tile of one (b, h) row.
// D = A x B with A = x1[16 w1 x 4 c], B = x2[4 c x 16 w2], f32 WMMA, K-loop over 128 channels.
__global__ __launch_bounds__(32) void corr1d_wmma_f32(const float* __restrict__ x1,
                                                      const float* __restrict__ x2,
                                                      float* __restrict__ out) {
    __shared__ float tile[ND * 16];   // [d][w_local] staging for coalesced stores

    const int wg = blockIdx.x;
    const int wt = wg % WTILES;
    const int h  = (wg / WTILES) % HDIM;
    const int b  = wg / (WTILES * HDIM);
    const int w1base = wt * 16;

    const int l    = threadIdx.x;     // 0..31 (wave32)
    const int half = l >> 4;          // lane group: 0 -> K={0,1}, 1 -> K={2,3}
    const int lm   = l & 15;          // M (for A) / N (for B) index
    const int kk   = half * 2;        // channel sub-offset within group of 4

    // ---- A operand: x1[b][kk + 4s][h][w1base + lm], second VGPR = next channel (+HWSZ)
    const float* pA = x1 + (((size_t)b * CDIM + kk) * HDIM + h) * (size_t)WDIM + w1base + lm;

    // ---- B operands: 4 tiles of shifted x2 columns; w2 fixed per lane across the K loop.
    // OOB lanes load from a clamped (valid) address and are zeroed by a float mask
    // (unconditional loads -> straight-line s_clause, mask mul co-executes with WMMA).
    // Tile 1 (w2 = w1base + lm) is always in-bounds: no mask needed.
    const float* pB0; const float* pB1; const float* pB2; const float* pB3;
    float m0, m2, m3;
    {
        const size_t rowoff = (((size_t)b * CDIM + kk) * HDIM + h) * (size_t)WDIM;
        int w2;
        w2 = w1base - 16 + lm;
        m0 = (w2 >= 0) ? 1.0f : 0.0f;
        pB0 = x2 + rowoff + (w2 < 0 ? 0 : w2);
        pB1 = x2 + rowoff + (w1base + lm);                    // always valid
        w2 = w1base + 16 + lm;
        m2 = (w2 < WDIM) ? 1.0f : 0.0f;
        pB2 = x2 + rowoff + (w2 >= WDIM ? WDIM - 1 : w2);
        w2 = w1base + 32 + lm;
        m3 = (w2 < WDIM) ? 1.0f : 0.0f;
        pB3 = x2 + rowoff + (w2 >= WDIM ? WDIM - 1 : w2);
    }

    v8f acc0 = {}, acc1 = {}, acc2 = {}, acc3 = {};

    #pragma unroll 4
    for (int s = 0; s < CDIM / 4; ++s) {
        v2f a;
        a.x = pA[0];
        a.y = pA[HWSZ];
        pA += 4 * HWSZ;

        // Unconditional loads (addresses pre-clamped), then mask.
        float t0a = pB0[0], t0b = pB0[HWSZ];  pB0 += 4 * HWSZ;
        float t1a = pB1[0], t1b = pB1[HWSZ];  pB1 += 4 * HWSZ;
        float t2a = pB2[0], t2b = pB2[HWSZ];  pB2 += 4 * HWSZ;
        float t3a = pB3[0], t3b = pB3[HWSZ];  pB3 += 4 * HWSZ;

        v2f b0, b1, b2, b3;
        b0.x = t0a * m0;  b0.y = t0b * m0;
        b1.x = t1a;       b1.y = t1b;
        b2.x = t2a * m2;  b2.y = t2b * m2;
        b3.x = t3a * m3;  b3.y = t3b * m3;

        // 8 args: (neg_a, A, neg_b, B, c_mod, C, reuse_a, reuse_b)
        acc0 = __builtin_amdgcn_wmma_f32_16x16x4_f32(false, a, false, b0, (short)0, acc0, false, false);
        acc1 = __builtin_amdgcn_wmma_f32_16x16x4_f32(false, a, false, b1, (short)0, acc1, false, false);
        acc2 = __builtin_amdgcn_wmma_f32_16x16x4_f32(false, a, false, b2, (short)0, acc2, false, false);
        acc3 = __builtin_amdgcn_wmma_f32_16x16x4_f32(false, a, false, b3, (short)0, acc3, false, false);
    }

    // ---- Scatter in-band entries to LDS: D tile t, lane entry (m = j + 8*half, n = lm),
    //      displacement d = 16*t + n - m - 8; exactly one t is in-band per (d, m).
    #pragma unroll
    for (int j = 0; j < 8; ++j) {
        const int m    = j + 8 * half;
        const int base = lm - m - 8;          // n - m - 8
        int d;
        d = base;      if (d >= 0 && d < ND) tile[d * 16 + m] = acc0[j];
        d = base + 16; if (d >= 0 && d < ND) tile[d * 16 + m] = acc1[j];
        d = base + 32; if (d >= 0 && d < ND) tile[d * 16 + m] = acc2[j];
        d = base + 48; if (d >= 0 && d < ND) tile[d * 16 + m] = acc3[j];
    }
    __syncthreads();

    // ---- Coalesced store: out[b][d][h][w1base + w]
    float* outp = out + (((size_t)b * ND) * HDIM + h) * (size_t)WDIM + w1base;
    for (int i = l; i < ND * 16; i += 32) {
        const int d = i >> 4;
        const int w = i & 15;
        outp[(size_t)d * HWSZ + w] = tile[i];
    }
}

extern "C" void kernel_launch(void* const* d_in, const int* in_sizes, int n_in,
                              void* d_out, int out_size, void* d_ws, size_t ws_size,
                              hipStream_t stream) {
    const float* x1 = (const float*)d_in[0];
    const float* x2 = (const float*)d_in[1];
    float* out = (float*)d_out;
    (void)in_sizes; (void)n_in; (void)out_size; (void)d_ws; (void)ws_size;

    const int grid = BDIM * HDIM * WTILES;   // 12800 workgroups, 1 wave32 each
    corr1d_wmma_f32<<<grid, 32, 0, stream>>>(x1, x2, out);
}